// MDRNN_42039139894313
// MI455X (gfx1250) — compile-verified
//
#include <hip/hip_runtime.h>

// MDRNN: h[i,j] = tanh(x[i,j]@w + bias + u0*h[i-1,j] + u1*h[i,j-1])
// D1=D2=128, B=16, SIN=64, SOUT=128.
//
// Kernel 1: WMMA f32 16x16x4 GEMM (a = x@w + bias), stores a transposed as
//           a_t[(b*SOUT+o)][i][j] via LDS tile for coalesced scan-side loads.
// Kernel 2: 2048 independent scalar 2D recurrences (u is elementwise!),
//           one wave32 per (b,o); diagonal wavefront inside the wave with
//           __shfl_up; strip boundaries through (same-wave, in-order) LDS.

#define DD1  128
#define DD2  128
#define BB   16
#define SIN  64
#define SOUT 128

typedef __attribute__((ext_vector_type(2))) float v2f;
typedef __attribute__((ext_vector_type(4))) float v4f;
typedef __attribute__((ext_vector_type(8))) float v8f;

// ---------------------------------------------------------------------------
// Kernel 1: per block: fixed i, 32 j's (j0..j0+31), 32 o's (o0..o0+31), all b.
// M = 32*16 = 512 rows (row m = jj*16 + b), N = 32, K = 64.
// 8 waves, each wave: 4 jj-tiles x 2 n-tiles = 8 WMMA tiles, 16 k-steps each.
// ---------------------------------------------------------------------------
__global__ __launch_bounds__(256)
void mdrnn_gemm_wmma(const float* __restrict__ x, const float* __restrict__ w,
                     const float* __restrict__ bias, float* __restrict__ a_t)
{
    __shared__ float tile[512 * 32];            // [m = jj*16+b][oo], 64 KB

    const int i    = blockIdx.z;
    const int j0   = blockIdx.y * 32;
    const int o0   = blockIdx.x * 32;
    const int tid  = threadIdx.x;
    const int lane = tid & 31;
    const int wave = tid >> 5;

    const int lm = lane & 15;                   // M (=b) / N column within tile
    const int kh = (lane >> 4) * 2;             // K sub-offset: lanes 16-31 -> K+2,K+3
    const int hh = lane >> 4;                   // D-row half select

    for (int t = 0; t < 8; ++t) {
        const int jj = wave * 4 + (t >> 1);     // local j (M-tile index)
        const int nn = (t & 1) * 16;            // N-tile offset within 32
        const int j  = j0 + jj;

        // A rows: m = jj*16 + b ; lane's row b = lm
        const float* xrow = x + (((size_t)i * DD2 + j) * BB + lm) * SIN;

        v8f acc = {};
        #pragma unroll
        for (int k0 = 0; k0 < SIN; k0 += 4) {
            // A 16x4 f32 frag: lanes 0-15 hold K=k0,k0+1; lanes 16-31 K=k0+2,k0+3
            v2f af;
            af.x = xrow[k0 + kh];
            af.y = xrow[k0 + kh + 1];
            // B 4x16 f32 frag (rows striped across lanes, mirrored K halves)
            const float* wp = w + (size_t)(k0 + kh) * SOUT + (o0 + nn + lm);
            v2f bf;
            bf.x = wp[0];
            bf.y = wp[SOUT];
            acc = __builtin_amdgcn_wmma_f32_16x16x4_f32(
                false, af, false, bf, (short)0, acc, false, false);
        }

        // D layout: lane, VGPR v -> M = v + 8*(lane>>4), N = lane&15
        #pragma unroll
        for (int v = 0; v < 8; ++v)
            tile[(jj * 16 + (v + 8 * hh)) * 32 + nn + lm] = acc[v];
    }
    __syncthreads();

    // Transposed store: a_t[(b*SOUT+o)][i][j0..j0+31], + bias.
    for (int p = tid; p < 512; p += 256) {
        const int b  = p >> 5;                  // 0..15
        const int oo = p & 31;                  // 0..31
        const int o  = o0 + oo;
        const float bv = bias[o];
        float* dst = a_t + (size_t)(b * SOUT + o) * (DD1 * DD2)
                         + (size_t)i * DD2 + j0;
        #pragma unroll
        for (int jj = 0; jj < 32; jj += 4) {
            v4f val;
            val.x = tile[((jj + 0) * 16 + b) * 32 + oo] + bv;
            val.y = tile[((jj + 1) * 16 + b) * 32 + oo] + bv;
            val.z = tile[((jj + 2) * 16 + b) * 32 + oo] + bv;
            val.w = tile[((jj + 3) * 16 + b) * 32 + oo] + bv;
            *(v4f*)(dst + jj) = val;
        }
    }
}

// ---------------------------------------------------------------------------
// Kernel 2: one wave per (b,o). Lanes = 32 columns j; 4 strips cover j=0..127.
// Diagonal time loop: at step t, lane l handles cell (i = t-l, j).
//   h[i][j-1] comes from lane l-1's value of step t-1 (shfl_up),
//   h[i-1][j]  is the lane-local value from step t-1,
//   strip-boundary column flows through per-wave LDS (same-wave => in-order).
// ---------------------------------------------------------------------------
__global__ __launch_bounds__(256)
void mdrnn_scan(const float* __restrict__ a_t, const float* __restrict__ u,
                float* __restrict__ h_out)
{
    __shared__ float boundary[8][DD1];          // per-wave strip boundary, 4 KB

    const int tid  = threadIdx.x;
    const int lane = tid & 31;
    const int wave = tid >> 5;
    const int bo   = blockIdx.x * 8 + wave;     // 0..2047
    const int b    = bo >> 7;
    const int o    = bo & 127;

    const float u0 = u[o];                      // u[0][o]
    const float u1 = u[SOUT + o];               // u[1][o]
    const float* abase = a_t + (size_t)bo * (DD1 * DD2);
    float* bnd = boundary[wave];

    for (int s = 0; s < 4; ++s) {
        const int j = s * 32 + lane;
        float h = 0.0f;                         // lane-local h[i-1][j] (0 at top)
        for (int t = 0; t < DD1 + 31; ++t) {
            float hl = __shfl_up(h, 1);         // h[i][j-1] from lane l-1 @ t-1
            const int i = t - lane;
            if (lane == 0)
                hl = (s > 0 && i >= 0 && i < DD1) ? bnd[i] : 0.0f;
            if (i >= 0 && i < DD1) {
                const float a  = abase[i * DD2 + j];    // one row i = t-l per lane
                const float hn = tanhf(fmaf(u0, h, fmaf(u1, hl, a)));
                h = hn;
                if (lane == 31) bnd[i] = hn;    // feed next strip's lane 0
                h_out[(((size_t)i * DD2 + j) * BB + b) * SOUT + o] = hn;
            }
        }
    }
}

extern "C" void kernel_launch(void* const* d_in, const int* in_sizes, int n_in,
                              void* d_out, int out_size, void* d_ws, size_t ws_size,
                              hipStream_t stream)
{
    const float* x    = (const float*)d_in[0];  // (128,128,16,64)
    const float* w    = (const float*)d_in[1];  // (64,128)
    const float* u    = (const float*)d_in[2];  // (2,128)
    const float* bias = (const float*)d_in[3];  // (128,)
    float* h   = (float*)d_out;                 // (128,128,16,128)
    float* a_t = (float*)d_ws;                  // 2048*16384 floats = 128 MB

    dim3 g1(SOUT / 32, DD2 / 32, DD1);          // (4, 4, 128)
    mdrnn_gemm_wmma<<<g1, 256, 0, stream>>>(x, w, bias, a_t);

    mdrnn_scan<<<(BB * SOUT) / 8, 256, 0, stream>>>(a_t, u, h);
}